// AdversarialGenerator_42812234006678
// MI455X (gfx1250) — compile-verified
//
#include <hip/hip_runtime.h>
#include <hip/hip_bf16.h>

#define NEGV -1000000000.0f
#define BB 4
#define LL 512
#define DD 64
#define HH 128

typedef __attribute__((ext_vector_type(2))) float v2f;
typedef __attribute__((ext_vector_type(8))) float v8f;

// ---------------------------------------------------------------------------
// Kernel 1: pi = X @ W1[:64], pj = X @ W1[64:]  via V_WMMA_F32_16X16X4_F32.
// One wave per 16x16 output tile. 2048 tiles total (2 outputs x 128 Mt x 8 Nt).
// A layout (16x4 f32): lanes 0-15 row M=lane, v0=K0 v1=K1; lanes 16-31 K2/K3.
// B layout (4x16 f32): lanes 0-15 col N=lane, v0=K0 v1=K1; lanes 16-31 K2/K3.
// C/D layout (16x16 f32): VGPR r -> row r (lanes 0-15) / row r+8 (lanes 16-31).
// A's two per-lane elements are contiguous -> one b64 load per k-step.
// ---------------------------------------------------------------------------
__global__ void gemm_pipj_wmma(const float* __restrict__ X,   // (2048, 64)
                               const float* __restrict__ W1,  // (128, 128)
                               float* __restrict__ pi,        // (2048, 128)
                               float* __restrict__ pj) {      // (2048, 128)
  int gtid  = blockIdx.x * blockDim.x + threadIdx.x;
  int wave  = gtid >> 5;
  int lane  = threadIdx.x & 31;
  int which = wave >> 10;          // 0 -> pi, 1 -> pj (1024 tiles each)
  int rem   = wave & 1023;
  int mt    = rem >> 3;            // 0..127
  int nt    = rem & 7;             // 0..7

  const float* W   = W1 + which * DD * HH;   // rows 0..63 or 64..127
  float*       OUT = which ? pj : pi;

  int row0 = mt * 16;
  int col0 = nt * 16;
  int hi   = (lane >> 4) & 1;      // lane half select
  int l15  = lane & 15;

  const float* arow = X + (size_t)(row0 + l15) * DD + 2 * hi;  // k0 + {0,2}
  const float* bcol = W + (size_t)col0 + l15 + (size_t)(2 * hi) * HH;

  v8f c = {};
#pragma unroll
  for (int kk = 0; kk < 16; ++kk) {
    int k0 = kk * 4;
    v2f a = *(const v2f*)(arow + k0);          // global_load_b64
    v2f b;
    b.x = bcol[(size_t)k0 * HH];
    b.y = bcol[(size_t)(k0 + 1) * HH];
    c = __builtin_amdgcn_wmma_f32_16x16x4_f32(false, a, false, b,
                                              (short)0, c, false, false);
  }
#pragma unroll
  for (int r = 0; r < 8; ++r) {
    int orow = row0 + r + (hi ? 8 : 0);
    OUT[orow * HH + col0 + l15] = c[r];
  }
}

// ---------------------------------------------------------------------------
// Kernel 2: la[b,i,j] = mask ? NEG : (sum_h relu(pi+pj+b1)*W2 + b2 + g)/tau
// Tile 16x16 (i x j) per 256-thread block. pi/pj tiles staged into LDS with
// GLOBAL_LOAD_ASYNC_TO_LDS_B128 (ASYNCcnt path, no VGPR round-trip), rows
// padded to 132 floats to kill stride-128 bank conflicts on the pj broadcast.
// ---------------------------------------------------------------------------
__global__ void scores_kernel(const float* __restrict__ pi,
                              const float* __restrict__ pj,
                              const float* __restrict__ b1,
                              const float* __restrict__ W2,   // (128,1) flat
                              const float* __restrict__ b2,   // (1,)
                              const float* __restrict__ gumbel,
                              const int*   __restrict__ seq_lens,
                              float* __restrict__ la) {
  int bidx = blockIdx.x;
  int jt = bidx & 31; bidx >>= 5;
  int it = bidx & 31; bidx >>= 5;
  int b  = bidx;

  __shared__ float spi[16][132];
  __shared__ float spj[16][132];
  __shared__ float sb1[HH];
  __shared__ float sw2[HH];

  int t = threadIdx.x;
  if (t < HH) { sb1[t] = b1[t]; sw2[t] = W2[t]; }

  const float* gpi = pi + (size_t)(b * LL + it * 16) * HH;
  const float* gpj = pj + (size_t)(b * LL + jt * 16) * HH;

  // 16 rows x 32 float4-chunks = 512 chunks per tile; 2 per thread.
#pragma unroll
  for (int pass = 0; pass < 2; ++pass) {
    int idx = t + pass * 256;
    int r = idx >> 5;
    int q = (idx & 31) * 4;                    // float column, multiple of 4
    unsigned lpi = (unsigned)(size_t)&spi[r][q];   // LDS byte offset
    unsigned lpj = (unsigned)(size_t)&spj[r][q];
    unsigned long long api = (unsigned long long)(size_t)(gpi + (size_t)r * HH + q);
    unsigned long long apj = (unsigned long long)(size_t)(gpj + (size_t)r * HH + q);
    asm volatile("global_load_async_to_lds_b128 %0, %1, off"
                 :: "v"(lpi), "v"(api) : "memory");
    asm volatile("global_load_async_to_lds_b128 %0, %1, off"
                 :: "v"(lpj), "v"(apj) : "memory");
  }
  asm volatile("s_wait_asynccnt 0x0" ::: "memory");
  __syncthreads();

  int ti = t >> 4, tj = t & 15;
  float s = 0.0f;
#pragma unroll 4
  for (int h = 0; h < HH; ++h) {
    float v = spi[ti][h] + spj[tj][h] + sb1[h];
    s = fmaf(fmaxf(v, 0.0f), sw2[h], s);
  }
  int i = it * 16 + ti, j = jt * 16 + tj;
  int sl = seq_lens[b];
  // tau = 0.5 -> multiply by 2
  float val = (s + b2[0] + gumbel[(b * LL + i) * LL + j]) * 2.0f;
  if (i >= sl || j >= sl) val = NEGV;
  la[(b * LL + i) * LL + j] = val;
}

// ---------------------------------------------------------------------------
// Kernel 3a: row logsumexp normalize (axis=2). One block per (b,i).
// ---------------------------------------------------------------------------
__global__ void row_lse_kernel(float* __restrict__ la) {
  float* p = la + (size_t)blockIdx.x * LL;
  int t = threadIdx.x;
  float x0 = p[t], x1 = p[t + 256];
  __shared__ float red[256];
  red[t] = fmaxf(x0, x1);
  __syncthreads();
  for (int o = 128; o > 0; o >>= 1) {
    if (t < o) red[t] = fmaxf(red[t], red[t + o]);
    __syncthreads();
  }
  float m = red[0];
  __syncthreads();
  red[t] = expf(x0 - m) + expf(x1 - m);
  __syncthreads();
  for (int o = 128; o > 0; o >>= 1) {
    if (t < o) red[t] += red[t + o];
    __syncthreads();
  }
  float lse = m + logf(red[0]);
  p[t] = x0 - lse;
  p[t + 256] = x1 - lse;
}

// ---------------------------------------------------------------------------
// Kernel 3b: column logsumexp normalize (axis=1) + re-apply pad mask.
// One block per (b,j).
// ---------------------------------------------------------------------------
__global__ void col_lse_kernel(float* __restrict__ la,
                               const int* __restrict__ seq_lens) {
  int col = blockIdx.x & (LL - 1);
  int b   = blockIdx.x >> 9;
  float* base = la + (size_t)b * LL * LL + col;
  int t = threadIdx.x;
  __builtin_prefetch(base + (size_t)(t + 64) * LL, 0, 0);   // global_prefetch_b8
  float x0 = base[(size_t)t * LL];
  float x1 = base[(size_t)(t + 256) * LL];
  __shared__ float red[256];
  red[t] = fmaxf(x0, x1);
  __syncthreads();
  for (int o = 128; o > 0; o >>= 1) {
    if (t < o) red[t] = fmaxf(red[t], red[t + o]);
    __syncthreads();
  }
  float m = red[0];
  __syncthreads();
  red[t] = expf(x0 - m) + expf(x1 - m);
  __syncthreads();
  for (int o = 128; o > 0; o >>= 1) {
    if (t < o) red[t] += red[t + o];
    __syncthreads();
  }
  float lse = m + logf(red[0]);
  int sl = seq_lens[b];
  float y0 = x0 - lse, y1 = x1 - lse;
  if (col >= sl) { y0 = NEGV; y1 = NEGV; }
  if (t >= sl)        y0 = NEGV;
  if (t + 256 >= sl)  y1 = NEGV;
  base[(size_t)t * LL]         = y0;
  base[(size_t)(t + 256) * LL] = y1;
}

// ---------------------------------------------------------------------------
// Kernel 4: perm = exp(la), written straight into d_out perm region.
// ---------------------------------------------------------------------------
__global__ void exp_kernel(const float* __restrict__ la,
                           float* __restrict__ perm_out) {
  int idx = blockIdx.x * 256 + threadIdx.x;
  perm_out[idx] = expf(la[idx]);
}

// ---------------------------------------------------------------------------
// Kernel 5: types/times_permed[b,m] = sum_l type/time[b,l] * perm[b,l,m]
// ---------------------------------------------------------------------------
__global__ void reduce_kernel(const float* __restrict__ perm,
                              const int*   __restrict__ etype,
                              const float* __restrict__ etime,
                              float* __restrict__ out_types,
                              float* __restrict__ out_times) {
  int m = blockIdx.x * 128 + threadIdx.x;
  int b = blockIdx.y;
  const float* pb = perm + (size_t)b * LL * LL;
  float st = 0.0f, sm = 0.0f;
  for (int l = 0; l < LL; ++l) {
    float p = pb[(size_t)l * LL + m];
    st = fmaf((float)etype[b * LL + l], p, st);
    sm = fmaf(etime[b * LL + l], p, sm);
  }
  out_types[b * LL + m] = st;
  out_times[b * LL + m] = sm;
}

extern "C" void kernel_launch(void* const* d_in, const int* in_sizes, int n_in,
                              void* d_out, int out_size, void* d_ws, size_t ws_size,
                              hipStream_t stream) {
  const float* event_time = (const float*)d_in[0];   // (B,L)
  const int*   event_type = (const int*)  d_in[1];   // (B,L)
  const float* x          = (const float*)d_in[2];   // (B,L,D)
  const int*   seq_lens   = (const int*)  d_in[3];   // (B,)
  const float* gumbel     = (const float*)d_in[4];   // (B,L,L)
  const float* W1         = (const float*)d_in[5];   // (2D,H)
  const float* b1         = (const float*)d_in[6];   // (H,)
  const float* W2         = (const float*)d_in[7];   // (H,1)
  const float* b2         = (const float*)d_in[8];   // (1,)

  float* out = (float*)d_out;
  float* out_types = out;                      // B*L
  float* out_times = out + BB * LL;            // B*L
  float* out_perm  = out + 2 * BB * LL;        // B*L*L

  // workspace layout
  float* pi = (float*)d_ws;                          // 2048*128
  float* pj = pi + (size_t)BB * LL * HH;             // 2048*128
  float* la = pj + (size_t)BB * LL * HH;             // 4*512*512

  // 1) pi/pj GEMMs: 2048 tiles, 4 waves/block -> 512 blocks x 128 threads
  gemm_pipj_wmma<<<512, 128, 0, stream>>>(x, W1, pi, pj);

  // 2) scores + gumbel + tau + mask -> la. B*32*32 = 4096 blocks
  scores_kernel<<<BB * 32 * 32, 256, 0, stream>>>(pi, pj, b1, W2, b2,
                                                  gumbel, seq_lens, la);

  // 3) 10 Sinkhorn iterations
  for (int it = 0; it < 10; ++it) {
    row_lse_kernel<<<BB * LL, 256, 0, stream>>>(la);
    col_lse_kernel<<<BB * LL, 256, 0, stream>>>(la, seq_lens);
  }

  // 4) perm = exp(la) into d_out
  exp_kernel<<<(BB * LL * LL) / 256, 256, 0, stream>>>(la, out_perm);

  // 5) weighted column sums
  reduce_kernel<<<dim3(LL / 128, BB), 128, 0, stream>>>(out_perm, event_type,
                                                        event_time,
                                                        out_types, out_times);
}